// BatchAllTripletLoss_1838246003075
// MI455X (gfx1250) — compile-verified
//
#include <hip/hip_runtime.h>
#include <hip/hip_bf16.h>

typedef __attribute__((ext_vector_type(2))) float v2f;
typedef __attribute__((ext_vector_type(8))) float v8f;

#define B_SZ 512
#define E_SZ 128
#define MARGIN_F 1.0f
#define EPS_F 1e-16f

// ---------------------------------------------------------------------------
// Kernel 1: per-row squared norms (diagonal of the Gram matrix).
// ---------------------------------------------------------------------------
__global__ void init_sqnorm_kernel(const float* __restrict__ embs,
                                   float* __restrict__ sqn) {
  const int i = blockIdx.x * blockDim.x + threadIdx.x;  // 0..511
  if (i < B_SZ) {
    const float4* row = (const float4*)(embs + (size_t)i * E_SZ);
    float s = 0.0f;
#pragma unroll
    for (int k = 0; k < E_SZ / 4; ++k) {
      const float4 v = row[k];
      s += v.x * v.x + v.y * v.y + v.z * v.z + v.w * v.w;
    }
    sqn[i] = s;
  }
}

// ---------------------------------------------------------------------------
// Kernel 2: dist[i][j] = max(sqn[i] + sqn[j] - 2 * <embs_i, embs_j>, 0)
// One wave32 per 16x16 output tile; K=128 via 32x V_WMMA_F32_16X16X4_F32.
//
// fp32 WMMA operand layout (16x16x4):
//   A (16x4, 2 VGPRs): lane = M + 16*(K>=2); VGPR0/1 = A[M][2*half + 0/1]
//   B (4x16, 2 VGPRs): lane = N + 16*half;   VGPR0/1 = B[2*half + 0/1][N]
// For a Gram matrix (B = A^T of the other tile) both reduce to the SAME
// per-lane pattern: embs[base + (lane&15)][k + 2*(lane>>4) + {0,1}].
//   C/D (16x16, 8 VGPRs): VGPR r, lane -> (M = r + 8*half, N = lane&15)
// ---------------------------------------------------------------------------
__global__ void dist_wmma_kernel(const float* __restrict__ embs,
                                 const float* __restrict__ sqn,
                                 float* __restrict__ dist) {
  const int tileM = blockIdx.x * 16;
  const int tileN = blockIdx.y * 16;
  const int lane  = threadIdx.x;   // 0..31, EXEC all ones (wave32)
  const int half  = lane >> 4;
  const int l     = lane & 15;

  const float* Arow = embs + (size_t)(tileM + l) * E_SZ + 2 * half;
  const float* Brow = embs + (size_t)(tileN + l) * E_SZ + 2 * half;

  v8f c = {};
#pragma unroll 8
  for (int k = 0; k < E_SZ; k += 4) {
    const v2f a = *(const v2f*)(Arow + k);
    const v2f b = *(const v2f*)(Brow + k);
    // 8 args: (neg_a, A, neg_b, B, c_mod, C, reuse_a, reuse_b)
    c = __builtin_amdgcn_wmma_f32_16x16x4_f32(false, a, false, b,
                                              (short)0, c, false, false);
  }

  const int   col   = tileN + l;
  const float sq_c  = sqn[col];
#pragma unroll
  for (int r = 0; r < 8; ++r) {
    const int row = tileM + r + 8 * half;
    const float d = sqn[row] + sq_c - 2.0f * c[r];
    dist[(size_t)row * B_SZ + col] = fmaxf(d, 0.0f);
  }
}

// ---------------------------------------------------------------------------
// Kernel 3: one block per anchor. Only valid triplets touched:
//   p: label[p]==label[a], p!=a        (~7 per anchor)
//   n: label[n]!=label[a]              (=> n!=a, n!=p automatically)
// Accumulate relu(d_ap - d_an + margin) and count(> EPS); fixed-order
// tree reduction -> per-anchor partials (deterministic, no float atomics).
// ---------------------------------------------------------------------------
__global__ void triplet_kernel(const float* __restrict__ dist,
                               const int* __restrict__ labels,
                               float* __restrict__ psum,
                               unsigned* __restrict__ pcnt) {
  __shared__ float    drow[B_SZ];
  __shared__ int      slab[B_SZ];
  __shared__ float    rsum[256];
  __shared__ unsigned rcnt[256];

  const int a   = blockIdx.x;
  const int tid = threadIdx.x;

  for (int i = tid; i < B_SZ; i += 256) {
    drow[i] = dist[(size_t)a * B_SZ + i];
    slab[i] = labels[i];
  }
  __syncthreads();

  const int lab_a = slab[a];
  float    sum = 0.0f;
  unsigned cnt = 0u;

  for (int p = 0; p < B_SZ; ++p) {
    if (p == a || slab[p] != lab_a) continue;
    const float dap_m = drow[p] + MARGIN_F;
    for (int n = tid; n < B_SZ; n += 256) {
      if (slab[n] == lab_a) continue;
      const float v = dap_m - drow[n];
      sum += fmaxf(v, 0.0f);          // relu contribution
      cnt += (v > EPS_F) ? 1u : 0u;   // n_pos: relu(v) > EPS  <=>  v > EPS
    }
  }

  rsum[tid] = sum;
  rcnt[tid] = cnt;
  __syncthreads();
#pragma unroll
  for (int s = 128; s > 0; s >>= 1) {
    if (tid < s) { rsum[tid] += rsum[tid + s]; rcnt[tid] += rcnt[tid + s]; }
    __syncthreads();
  }
  if (tid == 0) { psum[a] = rsum[0]; pcnt[a] = rcnt[0]; }
}

// ---------------------------------------------------------------------------
// Kernel 4: deterministic final reduction of 512 partials -> scalar.
// ---------------------------------------------------------------------------
__global__ void finalize_kernel(const float* __restrict__ psum,
                                const unsigned* __restrict__ pcnt,
                                float* __restrict__ out) {
  __shared__ float    rsum[B_SZ];
  __shared__ unsigned rcnt[B_SZ];
  const int tid = threadIdx.x;
  rsum[tid] = psum[tid];
  rcnt[tid] = pcnt[tid];
  __syncthreads();
#pragma unroll
  for (int s = 256; s > 0; s >>= 1) {
    if (tid < s) { rsum[tid] += rsum[tid + s]; rcnt[tid] += rcnt[tid + s]; }
    __syncthreads();
  }
  if (tid == 0) out[0] = rsum[0] / ((float)rcnt[0] + EPS_F);
}

// ---------------------------------------------------------------------------
extern "C" void kernel_launch(void* const* d_in, const int* in_sizes, int n_in,
                              void* d_out, int out_size, void* d_ws, size_t ws_size,
                              hipStream_t stream) {
  (void)in_sizes; (void)n_in; (void)out_size; (void)ws_size;

  const float* embs   = (const float*)d_in[0];
  const int*   labels = (const int*)d_in[1];
  float*       out    = (float*)d_out;

  // Workspace layout: dist (1 MB) | sqn (2 KB) | psum (2 KB) | pcnt (2 KB)
  char*     ws   = (char*)d_ws;
  float*    dist = (float*)ws;
  float*    sqn  = (float*)(ws + (size_t)B_SZ * B_SZ * sizeof(float));
  float*    psum = sqn + B_SZ;
  unsigned* pcnt = (unsigned*)(psum + B_SZ);

  init_sqnorm_kernel<<<dim3(2), dim3(256), 0, stream>>>(embs, sqn);
  dist_wmma_kernel<<<dim3(32, 32), dim3(32), 0, stream>>>(embs, sqn, dist);
  triplet_kernel<<<dim3(B_SZ), dim3(256), 0, stream>>>(dist, labels, psum, pcnt);
  finalize_kernel<<<dim3(1), dim3(B_SZ), 0, stream>>>(psum, pcnt, out);
}